// MMDLoss_23630910062680
// MI455X (gfx1250) — compile-verified
//
#include <hip/hip_runtime.h>
#include <hip/hip_bf16.h>

typedef float v2f __attribute__((ext_vector_type(2)));
typedef float v4f __attribute__((ext_vector_type(4)));
typedef float v8f __attribute__((ext_vector_type(8)));

#define NROWS 4096   // rows per input
#define DIMS  256    // feature dim
#define NTOT  8192   // 2*NROWS
#define KCHUNK 64    // K-chunk staged in LDS
#define BSTRIDE 68   // LDS row stride (floats); 68 % 64 == 4 -> conflict-free

// ---------------------------------------------------------------------------
// Kernel A: per-row sum of squares, sq[i] = sum_d T[i][d]^2, i in [0, 8192)
// ---------------------------------------------------------------------------
__global__ __launch_bounds__(256) void mmd_rowsq(const float* __restrict__ src,
                                                 const float* __restrict__ tgt,
                                                 float* __restrict__ sqv) {
  const int r = blockIdx.x * 256 + threadIdx.x;   // 0..8191 (32 blocks)
  const float* p = (r < NROWS) ? (src + (size_t)r * DIMS)
                               : (tgt + (size_t)(r - NROWS) * DIMS);
  float s = 0.0f;
  #pragma unroll 4
  for (int d = 0; d < DIMS; d += 4) {
    v4f v = *(const v4f*)(p + d);
    s += v.x * v.x + v.y * v.y + v.z * v.z + v.w * v.w;
  }
  sqv[r] = s;
}

// ---------------------------------------------------------------------------
// Kernel B (single block): bandwidth via the closed form
//   sum(L2) = 2n*sum(sq) - 2*||colsum||^2
// Stores gamma = log2(e) / (16 * bandwidth): the epilogue computes
// t = exp(-L2/(16 bw)) with one v_exp_f32; 5-kernel sum = t+t^2+t^4+t^8+t^16.
// ---------------------------------------------------------------------------
__global__ __launch_bounds__(256) void mmd_stats(const float* __restrict__ src,
                                                 const float* __restrict__ tgt,
                                                 const float* __restrict__ sqv,
                                                 float* __restrict__ gam) {
  __shared__ float red[256];
  const int tid = threadIdx.x;

  float c = 0.0f;
  for (int i = 0; i < NROWS; ++i) c += src[(size_t)i * DIMS + tid];
  for (int i = 0; i < NROWS; ++i) c += tgt[(size_t)i * DIMS + tid];
  red[tid] = c * c;
  __syncthreads();
  for (int s = 128; s > 0; s >>= 1) {
    if (tid < s) red[tid] += red[tid + s];
    __syncthreads();
  }
  const float csum2 = red[0];
  __syncthreads();

  float ss = 0.0f;
  for (int k = tid; k < NTOT; k += 256) ss += sqv[k];
  red[tid] = ss;
  __syncthreads();
  for (int s = 128; s > 0; s >>= 1) {
    if (tid < s) red[tid] += red[tid + s];
    __syncthreads();
  }

  if (tid == 0) {
    const double n = (double)NTOT;
    const double sumL2 = 2.0 * n * (double)red[0] - 2.0 * (double)csum2;
    double bw = sumL2 / (n * n - n);
    bw = bw / 4.0;  // / KERNEL_MUL^(KERNEL_NUM//2) = 2^2
    gam[0] = (float)(1.4426950408889634 / (16.0 * bw));  // log2(e)/(16*bw)
  }
}

// ---------------------------------------------------------------------------
// Kernel C: fused Gram-tile (f32 WMMA 16x16x4) + multi-bandwidth RBF epilogue.
// Grid 64x64, block = 256 threads = 8 waves; block tile = 128(M) x 128(N).
// B-panel is staged into LDS once per block (K-chunked, 34.8 KB, bank-
// conflict-free stride 68) so each element crosses the global/L2 path once;
// the 8 waves then feed B fragments via ds_load_b64.
// ---------------------------------------------------------------------------
__global__ __launch_bounds__(256) void mmd_wmma(const float* __restrict__ src,
                                                const float* __restrict__ tgt,
                                                const float* __restrict__ sqv,
                                                const float* __restrict__ gam,
                                                float* __restrict__ partial) {
  __shared__ float Bs[128 * BSTRIDE];   // 128 rows x 64-wide K chunk (padded)
  __shared__ float wsum[8];

  const int lane = threadIdx.x & 31;
  const int wave = threadIdx.x >> 5;
  const int mBlk = blockIdx.y * 128;          // global M base
  const int nBlk = blockIdx.x * 128;          // global N base
  const bool mSrc = mBlk < NROWS;
  const bool nSrc = nBlk < NROWS;
  const float* mBase = mSrc ? src : tgt;
  const float* nBase = nSrc ? src : tgt;
  const int mLoc = mSrc ? mBlk : (mBlk - NROWS);
  const int nLoc = nSrc ? nBlk : (nBlk - NROWS);

  const int halfK = (lane >> 4) << 1;         // 0 for lanes 0-15, 2 for 16-31
  const int r15   = lane & 15;

  const float* aPtr = mBase + (size_t)(mLoc + wave * 16 + r15) * DIMS + halfK;
  const float* bRow = nBase + (size_t)nLoc * DIMS;   // B panel base (global)
  const float* bL   = &Bs[r15 * BSTRIDE + halfK];    // B fragment base (LDS)

  v8f acc[8];
  #pragma unroll
  for (int t = 0; t < 8; ++t) acc[t] = (v8f){0.f,0.f,0.f,0.f,0.f,0.f,0.f,0.f};

  for (int kc = 0; kc < DIMS; kc += KCHUNK) {
    __syncthreads();   // previous chunk fully consumed
    // Stage B[0:128][kc:kc+64] -> LDS, b128 stores, coalesced b128 loads.
    #pragma unroll
    for (int idx = threadIdx.x; idx < 128 * (KCHUNK / 4); idx += 256) {
      const int r  = idx >> 4;          // row 0..127
      const int d4 = (idx & 15) << 2;   // 0,4,...,60
      *(v4f*)&Bs[r * BSTRIDE + d4] =
          *(const v4f*)(bRow + (size_t)r * DIMS + kc + d4);
    }
    __syncthreads();

    // 16 K-steps per chunk: 1 global A load + 8 LDS B loads + 8 WMMAs each.
    #pragma unroll 4
    for (int kk = 0; kk < KCHUNK; kk += 4) {
      v2f a = *(const v2f*)(aPtr + kc + kk);
      #pragma unroll
      for (int t = 0; t < 8; ++t) {
        v2f b = *(const v2f*)(bL + t * 16 * BSTRIDE + kk);
        acc[t] = __builtin_amdgcn_wmma_f32_16x16x4_f32(
            /*neg_a=*/false, a, /*neg_b=*/false, b,
            /*c_mod=*/(short)0, acc[t], /*reuse_a=*/false, /*reuse_b=*/false);
      }
    }
  }

  // Epilogue: L2 = sq[i] + sq[j] - 2*G; 5 RBF kernels via one exp.
  // C/D layout: VGPR r, lanes 0-15 -> (M=r, N=lane); lanes 16-31 -> (M=r+8).
  const float g = gam[0];
  const int rowG = mBlk + wave * 16 + ((lane >> 4) << 3);
  float sqI[8];
  #pragma unroll
  for (int r = 0; r < 8; ++r) sqI[r] = sqv[rowG + r];

  float s = 0.0f;
  #pragma unroll
  for (int t = 0; t < 8; ++t) {
    const float sqJ = sqv[nBlk + t * 16 + r15];
    #pragma unroll
    for (int r = 0; r < 8; ++r) {
      const float L2  = sqI[r] + sqJ - 2.0f * acc[t][r];
      const float tt  = __builtin_amdgcn_exp2f(-L2 * g);
      const float t2  = tt * tt;
      const float t4  = t2 * t2;
      const float t8  = t4 * t4;
      const float t16 = t8 * t8;
      s += tt + t2 + t4 + t8 + t16;
    }
  }

  // wave32 reduce, block reduce, one deterministic store per block
  #pragma unroll
  for (int off = 16; off > 0; off >>= 1) s += __shfl_xor(s, off, 32);

  if (lane == 0) wsum[wave] = s;
  __syncthreads();
  if (threadIdx.x == 0) {
    float b = 0.0f;
    #pragma unroll
    for (int w = 0; w < 8; ++w) b += wsum[w];
    const float sgn = (mSrc == nSrc) ? 1.0f : -1.0f;
    partial[blockIdx.y * gridDim.x + blockIdx.x] = sgn * b;
  }
}

// ---------------------------------------------------------------------------
// Kernel D: deterministic tree-reduce of 4096 block partials -> scalar MMD
// ---------------------------------------------------------------------------
__global__ __launch_bounds__(256) void mmd_final(const float* __restrict__ partial,
                                                 float* __restrict__ out) {
  __shared__ float red[256];
  const int tid = threadIdx.x;
  float s = 0.0f;
  for (int k = tid; k < 4096; k += 256) s += partial[k];
  red[tid] = s;
  __syncthreads();
  for (int st = 128; st > 0; st >>= 1) {
    if (tid < st) red[tid] += red[tid + st];
    __syncthreads();
  }
  if (tid == 0) out[0] = red[0] / (4096.0f * 4096.0f);
}

extern "C" void kernel_launch(void* const* d_in, const int* in_sizes, int n_in,
                              void* d_out, int out_size, void* d_ws, size_t ws_size,
                              hipStream_t stream) {
  const float* src = (const float*)d_in[0];   // [4096, 256] f32
  const float* tgt = (const float*)d_in[1];   // [4096, 256] f32
  float* wsf     = (float*)d_ws;
  float* sqv     = wsf;          // 8192 floats: row sums of squares
  float* gam     = wsf + 8192;   // 1 float: log2(e)/(16*bandwidth)
  float* partial = wsf + 8200;   // 4096 floats: per-block signed sums
  float* out     = (float*)d_out;

  mmd_rowsq<<<32, 256, 0, stream>>>(src, tgt, sqv);
  mmd_stats<<<1, 256, 0, stream>>>(src, tgt, sqv, gam);
  mmd_wmma<<<dim3(64, 64, 1), 256, 0, stream>>>(src, tgt, sqv, gam, partial);
  mmd_final<<<1, 256, 0, stream>>>(partial, out);
}